// ConvCaps_31894427140205
// MI455X (gfx1250) — compile-verified
//
#include <hip/hip_runtime.h>
#include <math.h>

// ---------------- problem constants ----------------
#define S_IN   32
#define S_OUT  30
#define C_INP  16
#define C_OUT  32
#define NCH    144          // 3*3*16 children per parent
#define P_TOT  900          // S_OUT^2
#define PPAD   904          // parent axis padded to mult of 4 (16B rows)
#define NC_TOT (NCH * C_OUT)   // 4608 (n,c) pairs
#define EPSV   1e-7f

typedef float v2f __attribute__((ext_vector_type(2)));
typedef float v8f __attribute__((ext_vector_type(8)));

// D = A(16x4 f32) x B(4x16 f32) + C, exact fp32 math (V_WMMA_F32_16X16X4_F32)
__device__ __forceinline__ v8f wmma4(v2f a, v2f b, v8f c) {
  return __builtin_amdgcn_wmma_f32_16x16x4_f32(false, a, false, b,
                                               (short)0, c, false, false);
}

// =====================================================================
// M-step: one wave per 4-parent quad. Streams all 144 children through
// WMMA, accumulating Sum(rr), Sum(rr*v), Sum(rr*v^2) in registers.
// UNIFORM/FINAL are compile-time to keep the hot loop branch-free.
// =====================================================================
template <bool UNIFORM, bool FINAL>
__global__ __launch_bounds__(32)
void m_step_kernel(const float* __restrict__ pose,      // [32][32][16][4][4]
                   const float* __restrict__ W,         // [144][32][4][4]
                   const float* __restrict__ rr,        // [144*32][PPAD]
                   const float* __restrict__ beta_a,    // [32]
                   const float* __restrict__ beta_v,    // [32]
                   float* __restrict__ means_out,       // [900][32][16]
                   float* __restrict__ i2v_out,         // [900][32][16]
                   float* __restrict__ apc_out,         // [900][32]
                   float* __restrict__ act_out,         // [900][32]
                   float lambd)
{
  const int quad = blockIdx.x;        // 0..224  (4 parents each)
  const int lane = threadIdx.x;       // 0..31
  const int half = lane >> 4;
  const int l    = lane & 15;
  // B / D column role: col N = p_l*4 + i
  const int p_l = l >> 2;
  const int i4  = l & 3;
  const int k0  = half * 2;           // this lane's K pair for A and B
  // A row role: row M = c_l*4 + j
  const int cA  = l >> 2;
  const int jA  = l & 3;
  const int p   = quad * 4 + p_l;
  const int py  = p / S_OUT, px = p - py * S_OUT;
  const int cD  = 2 * half;           // D rows r<4 -> c=ct*4+cD, r>=4 -> +1

  float acc1[8][8], acc2[8][8], rsA[8][2];
#pragma unroll
  for (int t = 0; t < 8; ++t) {
    rsA[t][0] = 0.f; rsA[t][1] = 0.f;
#pragma unroll
    for (int r = 0; r < 8; ++r) { acc1[t][r] = 0.f; acc2[t][r] = 0.f; }
  }

  for (int n = 0; n < NCH; ++n) {
    const int kk = n >> 4, ci = n & 15;
    const int ky = kk / 3, kx = kk - ky * 3;
    const int sy = py + ky, sx = px + kx;
    const float* pp = pose + (((sy * S_IN + sx) * C_INP + ci) * 16 + i4 * 4 + k0);
    v2f b; b.x = pp[0]; b.y = pp[1];                       // b64 load
    if (n + 1 < NCH)
      __builtin_prefetch(W + (size_t)(n + 1) * C_OUT * 16, 0, 1);

    // ---- issue all A fragments (and rr pairs) as load clauses ----
    v2f a[8];
#pragma unroll
    for (int t = 0; t < 8; ++t) {
      const float* wp = W + ((n * C_OUT + t * 4 + cA) * 16 + jA);
      a[t].x = wp[k0 * 4];
      a[t].y = wp[(k0 + 1) * 4];
    }
    float r0[8], r1[8];
#pragma unroll
    for (int t = 0; t < 8; ++t) {
      if (UNIFORM) { r0[t] = 1.0f / 32.0f; r1[t] = 1.0f / 32.0f; }
      else {
        const int c0 = t * 4 + cD;
        r0[t] = rr[(n * C_OUT + c0)     * PPAD + p];
        r1[t] = rr[(n * C_OUT + c0 + 1) * PPAD + p];
      }
    }

    // ---- 8 independent WMMAs + branch-free accumulation ----
#pragma unroll
    for (int t = 0; t < 8; ++t) {
      v8f c = {};
      v8f d = wmma4(a[t], b, c);                           // votes tile
      rsA[t][0] += r0[t]; rsA[t][1] += r1[t];
#pragma unroll
      for (int r = 0; r < 8; ++r) {
        float rv = (r < 4) ? r0[t] : r1[t];
        float v  = d[r];
        acc1[t][r] += rv * v;
        acc2[t][r] += rv * v * v;
      }
    }
  }

#pragma unroll
  for (int t = 0; t < 8; ++t) {
    const int c0 = t * 4 + cD;
    const float rs0 = rsA[t][0], rs1 = rsA[t][1];
    float mean[8], iv[8];
    float slog0 = 0.f, slog1 = 0.f;
#pragma unroll
    for (int r = 0; r < 8; ++r) {
      float rs  = (r < 4) ? rs0 : rs1;
      float m   = acc1[t][r] / rs;
      float var = acc2[t][r] / rs - m * m;
      var = fmaxf(var, 1e-30f);
      float sd  = sqrtf(var);
      float lst = logf(sd + EPSV);
      mean[r] = m;
      iv[r]   = 0.5f / var;             // 1/(2*std^2)
      if (r < 4) slog0 += lst; else slog1 += lst;
    }
    // sum over the i-component (lane bits 0..1) -> Sum_e log(std+eps)
    slog0 += __shfl_xor(slog0, 1, 32); slog0 += __shfl_xor(slog0, 2, 32);
    slog1 += __shfl_xor(slog1, 1, 32); slog1 += __shfl_xor(slog1, 2, 32);
    const float sc0 = 16.f * beta_v[c0]     + rs0 * slog0;
    const float sc1 = 16.f * beta_v[c0 + 1] + rs1 * slog1;
    const float a0 = 1.f / (1.f + expf(-lambd * (beta_a[c0]     - sc0)));
    const float a1 = 1.f / (1.f + expf(-lambd * (beta_a[c0 + 1] - sc1)));

    float4 m0 = make_float4(mean[0], mean[1], mean[2], mean[3]);
    float4 m1 = make_float4(mean[4], mean[5], mean[6], mean[7]);
    *(float4*)(means_out + ((p * C_OUT + c0)     * 16 + i4 * 4)) = m0;
    *(float4*)(means_out + ((p * C_OUT + c0 + 1) * 16 + i4 * 4)) = m1;
    if (!FINAL) {
      float4 w0 = make_float4(iv[0], iv[1], iv[2], iv[3]);
      float4 w1 = make_float4(iv[4], iv[5], iv[6], iv[7]);
      *(float4*)(i2v_out + ((p * C_OUT + c0)     * 16 + i4 * 4)) = w0;
      *(float4*)(i2v_out + ((p * C_OUT + c0 + 1) * 16 + i4 * 4)) = w1;
      if (i4 == 0) {
        apc_out[p * C_OUT + c0]     = logf(a0 + EPSV) - slog0;
        apc_out[p * C_OUT + c0 + 1] = logf(a1 + EPSV) - slog1;
      }
    } else {
      if (i4 == 0) {
        act_out[p * C_OUT + c0]     = a0;
        act_out[p * C_OUT + c0 + 1] = a1;
      }
    }
  }
}

// =====================================================================
// E-step part 1: recompute votes (WMMA) and write
//   zz[n][c][p] = apc[p,c] - Sum_e (v-mean)^2 * i2v
// =====================================================================
__global__ __launch_bounds__(32)
void e_zz_kernel(const float* __restrict__ pose,
                 const float* __restrict__ W,
                 const float* __restrict__ means,
                 const float* __restrict__ i2v,
                 const float* __restrict__ apc,
                 float* __restrict__ zz)          // [144*32][PPAD]
{
  const int quad = blockIdx.x;
  const int lane = threadIdx.x;
  const int half = lane >> 4;
  const int l    = lane & 15;
  const int p_l = l >> 2, i4 = l & 3, k0 = half * 2;
  const int cA  = l >> 2, jA = l & 3;
  const int p   = quad * 4 + p_l;
  const int py  = p / S_OUT, px = p - py * S_OUT;
  const int cD  = 2 * half;

  float m[8][8], iv[8][8], ap0[8], ap1[8];
#pragma unroll
  for (int t = 0; t < 8; ++t) {
    const int c0 = t * 4 + cD;
    float4 x0 = *(const float4*)(means + ((p * C_OUT + c0)     * 16 + i4 * 4));
    float4 x1 = *(const float4*)(means + ((p * C_OUT + c0 + 1) * 16 + i4 * 4));
    m[t][0] = x0.x; m[t][1] = x0.y; m[t][2] = x0.z; m[t][3] = x0.w;
    m[t][4] = x1.x; m[t][5] = x1.y; m[t][6] = x1.z; m[t][7] = x1.w;
    float4 y0 = *(const float4*)(i2v + ((p * C_OUT + c0)     * 16 + i4 * 4));
    float4 y1 = *(const float4*)(i2v + ((p * C_OUT + c0 + 1) * 16 + i4 * 4));
    iv[t][0] = y0.x; iv[t][1] = y0.y; iv[t][2] = y0.z; iv[t][3] = y0.w;
    iv[t][4] = y1.x; iv[t][5] = y1.y; iv[t][6] = y1.z; iv[t][7] = y1.w;
    ap0[t] = apc[p * C_OUT + c0];
    ap1[t] = apc[p * C_OUT + c0 + 1];
  }

  for (int n = 0; n < NCH; ++n) {
    const int kk = n >> 4, ci = n & 15;
    const int ky = kk / 3, kx = kk - ky * 3;
    const int sy = py + ky, sx = px + kx;
    const float* pp = pose + (((sy * S_IN + sx) * C_INP + ci) * 16 + i4 * 4 + k0);
    v2f b; b.x = pp[0]; b.y = pp[1];

    v2f a[8];
#pragma unroll
    for (int t = 0; t < 8; ++t) {
      const float* wp = W + ((n * C_OUT + t * 4 + cA) * 16 + jA);
      a[t].x = wp[k0 * 4];
      a[t].y = wp[(k0 + 1) * 4];
    }

#pragma unroll
    for (int t = 0; t < 8; ++t) {
      v8f c = {};
      v8f d = wmma4(a[t], b, c);
      float s0 = 0.f, s1 = 0.f;
#pragma unroll
      for (int r = 0; r < 8; ++r) {
        float dv = d[r] - m[t][r];
        float q  = dv * dv * iv[t][r];
        if (r < 4) s0 += q; else s1 += q;
      }
      s0 += __shfl_xor(s0, 1, 32); s0 += __shfl_xor(s0, 2, 32);
      s1 += __shfl_xor(s1, 1, 32); s1 += __shfl_xor(s1, 2, 32);
      if (i4 == 0) {
        const int c0 = t * 4 + cD;
        zz[(n * C_OUT + c0)     * PPAD + p] = ap0[t] - s0;
        zz[(n * C_OUT + c0 + 1) * PPAD + p] = ap1[t] - s1;
      }
    }
  }
}

// =====================================================================
// E-step part 2: per (n,c) softmax statistics over the parent axis.
// =====================================================================
__global__ __launch_bounds__(256)
void e_reduce_kernel(const float* __restrict__ zz, float* __restrict__ red)
{
  __shared__ float buf[P_TOT];
  __shared__ float tmp[256];
  const int nc = blockIdx.x, tid = threadIdx.x;
  const float* row = zz + (size_t)nc * PPAD;
  float mx = -3.4e38f;
  for (int p = tid; p < P_TOT; p += 256) {
    float v = row[p]; buf[p] = v; mx = fmaxf(mx, v);
  }
  tmp[tid] = mx; __syncthreads();
  for (int s = 128; s > 0; s >>= 1) {
    if (tid < s) tmp[tid] = fmaxf(tmp[tid], tmp[tid + s]);
    __syncthreads();
  }
  mx = tmp[0]; __syncthreads();
  float sm = 0.f;
  for (int p = tid; p < P_TOT; p += 256) sm += expf(buf[p] - mx);
  tmp[tid] = sm; __syncthreads();
  for (int s = 128; s > 0; s >>= 1) {
    if (tid < s) tmp[tid] += tmp[tid + s];
    __syncthreads();
  }
  if (tid == 0) { red[nc * 2] = mx; red[nc * 2 + 1] = tmp[0]; }
}

// E-step part 3: zz -> rr in place.
__global__ __launch_bounds__(256)
void e_norm_kernel(float* __restrict__ zz, const float* __restrict__ red)
{
  const int idx = blockIdx.x * 256 + threadIdx.x;
  if (idx >= NC_TOT * PPAD) return;
  const int nc = idx / PPAD;
  const int p  = idx - nc * PPAD;
  float v = 0.f;
  if (p < P_TOT) v = expf(zz[idx] - red[nc * 2]) / red[nc * 2 + 1];
  zz[idx] = v;
}

// =====================================================================
extern "C" void kernel_launch(void* const* d_in, const int* in_sizes, int n_in,
                              void* d_out, int out_size, void* d_ws, size_t ws_size,
                              hipStream_t stream) {
  (void)in_sizes; (void)n_in; (void)out_size; (void)ws_size;
  const float* pose   = (const float*)d_in[1];   // d_in[0] (input_act) unused by ref math
  const float* W      = (const float*)d_in[2];
  const float* beta_a = (const float*)d_in[3];
  const float* beta_v = (const float*)d_in[4];
  float* out = (float*)d_out;                    // [900*32] act ++ [900*32*16] pose

  float* ws     = (float*)d_ws;                  // ~20.5 MB total
  float* rrzz   = ws;                            // [4608][904]
  float* meansW = rrzz + (size_t)NC_TOT * PPAD;  // [900][32][16]
  float* i2vW   = meansW + (size_t)P_TOT * C_OUT * 16;
  float* apcW   = i2vW  + (size_t)P_TOT * C_OUT * 16;
  float* redW   = apcW  + (size_t)P_TOT * C_OUT; // [4608][2]

  const float lam0 = 0.0f;
  const float lam1 = 0.01f * (1.0f - 0.95f);
  const float lam2 = 0.01f * (1.0f - 0.95f * 0.95f);

  dim3 gq(225), bq(32);
  const int normBlocks = (NC_TOT * PPAD + 255) / 256;

  // ---- iteration 0 (uniform rr) ----
  m_step_kernel<true, false><<<gq, bq, 0, stream>>>(pose, W, rrzz, beta_a, beta_v,
                                                    meansW, i2vW, apcW, out, lam0);
  e_zz_kernel<<<gq, bq, 0, stream>>>(pose, W, meansW, i2vW, apcW, rrzz);
  e_reduce_kernel<<<NC_TOT, 256, 0, stream>>>(rrzz, redW);
  e_norm_kernel<<<normBlocks, 256, 0, stream>>>(rrzz, redW);

  // ---- iteration 1 ----
  m_step_kernel<false, false><<<gq, bq, 0, stream>>>(pose, W, rrzz, beta_a, beta_v,
                                                     meansW, i2vW, apcW, out, lam1);
  e_zz_kernel<<<gq, bq, 0, stream>>>(pose, W, meansW, i2vW, apcW, rrzz);
  e_reduce_kernel<<<NC_TOT, 256, 0, stream>>>(rrzz, redW);
  e_norm_kernel<<<normBlocks, 256, 0, stream>>>(rrzz, redW);

  // ---- iteration 2 (final): write act + pose straight into d_out ----
  m_step_kernel<false, true><<<gq, bq, 0, stream>>>(pose, W, rrzz, beta_a, beta_v,
                                                    out + P_TOT * C_OUT,   // pose region
                                                    i2vW, apcW,
                                                    out,                   // act region
                                                    lam2);
}